// STGCN_Encoder_32555852104111
// MI455X (gfx1250) — compile-verified
//
#include <hip/hip_runtime.h>
#include <hip/hip_bf16.h>

// ST-GCN encoder for MI455X (gfx1250, wave32).
// All GEMM-shaped math on V_WMMA_F32_16X16X4_F32 (exact fp32 MACs).
// Round-1 fix: weights are pre-padded/pre-scaled into ws (prep kernels) so
// A-fragment loads are unconditional b64 loads; TCN weights are reordered
// tap-major (kt*cin+i) so the im2col offset and boundary predicate are
// loop-invariant per tap and B loads strength-reduce to ptr += 4*S with
// branchless v_cndmask zeroing at the t edges.

#define EPSBN 1e-5f
#define T_LEN 512
#define V_NOD 25
#define N_BAT 32
#define S_TOT (N_BAT * T_LEN * V_NOD)   // 409600, divisible by 64

typedef float v2f __attribute__((ext_vector_type(2)));
typedef float v8f __attribute__((ext_vector_type(8)));

// ---------------------------------------------------------------------------
__global__ void __launch_bounds__(256)
k_databn(const float* __restrict__ x,
         const float* __restrict__ g, const float* __restrict__ b,
         const float* __restrict__ m, const float* __restrict__ vv,
         float* __restrict__ H)
{
    int idx = blockIdx.x * blockDim.x + threadIdx.x;
    const int total = N_BAT * 3 * T_LEN * V_NOD;
    if (idx >= total) return;
    int v = idx % V_NOD;
    int t = (idx / V_NOD) % T_LEN;
    int c = (idx / (V_NOD * T_LEN)) % 3;
    int n = idx / (V_NOD * T_LEN * 3);
    int ch = v * 3 + c;
    float sc = g[ch] * rsqrtf(vv[ch] + EPSBN);
    float sh = b[ch] - m[ch] * sc;
    size_t o = ((size_t)c * N_BAT + n) * (T_LEN * V_NOD) + t * V_NOD + v;
    H[o] = x[idx] * sc + sh;
}

__global__ void k_colsum(const float* __restrict__ A, float* __restrict__ As)
{
    int w = threadIdx.x;
    if (w < V_NOD) {
        float s = 0.f;
        for (int v = 0; v < V_NOD; ++v) s += A[v * V_NOD + w];
        As[w] = s;
    }
}

__global__ void __launch_bounds__(256)
k_amix(const float* __restrict__ H, const float* __restrict__ A,
       float* __restrict__ Tmp, int C)
{
    __shared__ float sA[V_NOD * V_NOD];
    for (int i = threadIdx.x; i < V_NOD * V_NOD; i += blockDim.x) sA[i] = A[i];
    __syncthreads();
    size_t idx = (size_t)blockIdx.x * blockDim.x + threadIdx.x;
    size_t total = (size_t)C * S_TOT;
    if (idx >= total) return;
    int w = (int)(idx % V_NOD);
    size_t base = idx - w;
    float s = 0.f;
#pragma unroll
    for (int v = 0; v < V_NOD; ++v) s += H[base + v] * sA[v * V_NOD + w];
    Tmp[idx] = s;
}

// ---------------------------------------------------------------------------
// weight prep: pad gcn weights to Kup columns (zeros beyond K)
__global__ void __launch_bounds__(256)
k_prep_gw(const float* __restrict__ gw, float* __restrict__ Wg,
          int co, int K, int Kup)
{
    int idx = blockIdx.x * blockDim.x + threadIdx.x;
    if (idx >= co * Kup) return;
    int o = idx / Kup, k = idx % Kup;
    Wg[idx] = (k < K) ? gw[o * K + k] : 0.f;
}

// weight prep: tcn weights reordered tap-major and scaled by bn2; residual
// 1x1 weights appended (scaled by residual-BN); zero padding to KupT.
__global__ void __launch_bounds__(256)
k_prep_tw(const float* __restrict__ tw, const float* __restrict__ rw,
          const float* __restrict__ b2g, const float* __restrict__ b2v,
          const float* __restrict__ rbg, const float* __restrict__ rbv,
          float* __restrict__ Wt, int co, int cin, int cir, int KupT,
          int hasRes)
{
    int idx = blockIdx.x * blockDim.x + threadIdx.x;
    if (idx >= co * KupT) return;
    int o = idx / KupT, r = idx % KupT;
    int K1 = cin * 9;
    float w = 0.f;
    if (r < K1) {
        int kt = r / cin, i = r - kt * cin;          // tap-major
        float s2 = b2g[o] * rsqrtf(b2v[o] + EPSBN);
        w = tw[((size_t)o * cin + i) * 9 + kt] * s2;
    } else if (hasRes && r < K1 + cir) {
        float sR = rbg[o] * rsqrtf(rbv[o] + EPSBN);
        w = rw[(size_t)o * cir + (r - K1)] * sR;
    }
    Wt[idx] = w;
}

// ---------------------------------------------------------------------------
// GCN GEMM: Out[o,s] = relu( bn1( sum_c Wg[o,c]*Tmp[c,s] + gb[o]*As[s%25] ) )
__global__ void __launch_bounds__(256)
k_gcn(const float* __restrict__ Wg,    // co x Kup (padded)
      const float* __restrict__ Bact,  // rows x S_TOT (ws buffer, has slack)
      const float* __restrict__ gb,
      const float* __restrict__ b1g, const float* __restrict__ b1b,
      const float* __restrict__ b1m, const float* __restrict__ b1v,
      const float* __restrict__ As,
      float* __restrict__ Out, int co, int Kup)
{
    const int lane = threadIdx.x & 31;
    const int half = lane >> 4, lq = lane & 15;
    const int wid  = blockIdx.x * 8 + (threadIdx.x >> 5);
    const int ntN  = S_TOT >> 6;                 // 6400
    const int m0 = (wid / ntN) << 6;
    const int n0 = (wid % ntN) << 6;
    if (m0 >= co) return;                         // wave-uniform

    const float* Ar[4];
#pragma unroll
    for (int mi = 0; mi < 4; ++mi)
        Ar[mi] = Wg + (size_t)(m0 + 16 * mi + lq) * Kup + 2 * half;
    const float* Bp[4];
#pragma unroll
    for (int ni = 0; ni < 4; ++ni)
        Bp[ni] = Bact + (size_t)(2 * half) * S_TOT + (n0 + 16 * ni + lq);

    const v8f vz = {0.f,0.f,0.f,0.f,0.f,0.f,0.f,0.f};
    v8f acc[4][4];
#pragma unroll
    for (int i = 0; i < 4; ++i)
#pragma unroll
        for (int j = 0; j < 4; ++j) acc[i][j] = vz;

    for (int kk = 0; kk < Kup; kk += 4) {
        v2f a[4], b[4];
#pragma unroll
        for (int mi = 0; mi < 4; ++mi)
            a[mi] = *(const v2f*)(Ar[mi] + kk);
#pragma unroll
        for (int ni = 0; ni < 4; ++ni) {
            b[ni].x = Bp[ni][0];
            b[ni].y = Bp[ni][S_TOT];
            Bp[ni] += 4 * (size_t)S_TOT;
        }
#pragma unroll
        for (int mi = 0; mi < 4; ++mi)
#pragma unroll
            for (int ni = 0; ni < 4; ++ni)
                acc[mi][ni] = __builtin_amdgcn_wmma_f32_16x16x4_f32(
                    false, a[mi], false, b[ni], (short)0, acc[mi][ni],
                    false, false);
    }
#pragma unroll
    for (int mi = 0; mi < 4; ++mi) {
#pragma unroll
        for (int r = 0; r < 8; ++r) {
            int o = m0 + 16 * mi + r + 8 * half;
            float s1  = b1g[o] * rsqrtf(b1v[o] + EPSBN);
            float sh1 = b1b[o] - b1m[o] * s1;
            float gbo = gb[o];
#pragma unroll
            for (int ni = 0; ni < 4; ++ni) {
                int col = n0 + 16 * ni + lq;
                float val = acc[mi][ni][r] + gbo * As[col % V_NOD];
                val = val * s1 + sh1;
                Out[(size_t)o * S_TOT + col] = fmaxf(val, 0.f);
            }
        }
    }
}

// ---------------------------------------------------------------------------
// TCN GEMM, tap-major K: phase 1 loops 9 taps x (cin/4) K-steps with
// loop-invariant dt offset + branchless edge zeroing; phase 2 adds the
// fused residual 1x1 rows. Epilogue: bn2 + residual consts (+identity) +relu.
__global__ void __launch_bounds__(256)
k_tcn(const float* __restrict__ Wt,    // co x KupT (prepped, scaled)
      const float* __restrict__ G,     // cin x S_TOT (ws, slack both sides)
      const float* __restrict__ Hin,   // residual source rows x S_TOT
      const float* __restrict__ tb,
      const float* __restrict__ b2g, const float* __restrict__ b2b,
      const float* __restrict__ b2m, const float* __restrict__ b2v,
      const float* __restrict__ rb,
      const float* __restrict__ rbg, const float* __restrict__ rbb,
      const float* __restrict__ rbm, const float* __restrict__ rbv,
      float* __restrict__ Out, int co, int cin, int KupT, int K2p,
      int hasRes, int hasIdent)
{
    const int lane = threadIdx.x & 31;
    const int half = lane >> 4, lq = lane & 15;
    const int wid  = blockIdx.x * 8 + (threadIdx.x >> 5);
    const int ntN  = S_TOT >> 6;
    const int m0 = (wid / ntN) << 6;
    const int n0 = (wid % ntN) << 6;
    if (m0 >= co) return;

    const float* Ar[4];
#pragma unroll
    for (int mi = 0; mi < 4; ++mi)
        Ar[mi] = Wt + (size_t)(m0 + 16 * mi + lq) * KupT + 2 * half;

    int col[4], tcol[4];
#pragma unroll
    for (int ni = 0; ni < 4; ++ni) {
        col[ni]  = n0 + 16 * ni + lq;
        tcol[ni] = (col[ni] / V_NOD) & (T_LEN - 1);
    }

    const v8f vz = {0.f,0.f,0.f,0.f,0.f,0.f,0.f,0.f};
    v8f acc[4][4];
#pragma unroll
    for (int i = 0; i < 4; ++i)
#pragma unroll
        for (int j = 0; j < 4; ++j) acc[i][j] = vz;

    // phase 1: temporal taps
    for (int kt = 0; kt < 9; ++kt) {
        const int dt = kt - 4;
        const float* q[4];
        bool val[4];
#pragma unroll
        for (int ni = 0; ni < 4; ++ni) {
            val[ni] = ((unsigned)(tcol[ni] + dt)) < (unsigned)T_LEN;
            q[ni] = G + (size_t)(2 * half) * S_TOT + (col[ni] + dt * V_NOD);
        }
        const int kb = kt * cin;
        for (int ii = 0; ii < cin; ii += 4) {
            v2f a[4], b[4];
#pragma unroll
            for (int mi = 0; mi < 4; ++mi)
                a[mi] = *(const v2f*)(Ar[mi] + kb + ii);
#pragma unroll
            for (int ni = 0; ni < 4; ++ni) {
                float x = q[ni][0];          // unconditional load (ws slack)
                float y = q[ni][S_TOT];
                b[ni].x = val[ni] ? x : 0.f; // v_cndmask, no exec dance
                b[ni].y = val[ni] ? y : 0.f;
                q[ni] += 4 * (size_t)S_TOT;
            }
#pragma unroll
            for (int mi = 0; mi < 4; ++mi)
#pragma unroll
                for (int ni = 0; ni < 4; ++ni)
                    acc[mi][ni] = __builtin_amdgcn_wmma_f32_16x16x4_f32(
                        false, a[mi], false, b[ni], (short)0, acc[mi][ni],
                        false, false);
        }
    }
    // phase 2: fused residual 1x1 rows (weights pre-scaled; pads are zero)
    if (K2p > 0) {
        const int kb = cin * 9;
        const float* q[4];
#pragma unroll
        for (int ni = 0; ni < 4; ++ni)
            q[ni] = Hin + (size_t)(2 * half) * S_TOT + col[ni];
        for (int ii = 0; ii < K2p; ii += 4) {
            v2f a[4], b[4];
#pragma unroll
            for (int mi = 0; mi < 4; ++mi)
                a[mi] = *(const v2f*)(Ar[mi] + kb + ii);
#pragma unroll
            for (int ni = 0; ni < 4; ++ni) {
                b[ni].x = q[ni][0];
                b[ni].y = q[ni][S_TOT];
                q[ni] += 4 * (size_t)S_TOT;
            }
#pragma unroll
            for (int mi = 0; mi < 4; ++mi)
#pragma unroll
                for (int ni = 0; ni < 4; ++ni)
                    acc[mi][ni] = __builtin_amdgcn_wmma_f32_16x16x4_f32(
                        false, a[mi], false, b[ni], (short)0, acc[mi][ni],
                        false, false);
        }
    }
#pragma unroll
    for (int mi = 0; mi < 4; ++mi) {
#pragma unroll
        for (int r = 0; r < 8; ++r) {
            int o = m0 + 16 * mi + r + 8 * half;
            float s2 = b2g[o] * rsqrtf(b2v[o] + EPSBN);
            float c0 = tb[o] * s2 + (b2b[o] - b2m[o] * s2);
            if (hasRes) {
                float sR = rbg[o] * rsqrtf(rbv[o] + EPSBN);
                c0 += rb[o] * sR + (rbb[o] - rbm[o] * sR);
            }
#pragma unroll
            for (int ni = 0; ni < 4; ++ni) {
                float val = acc[mi][ni][r] + c0;
                if (hasIdent) val += Hin[(size_t)o * S_TOT + col[ni]];
                Out[(size_t)o * S_TOT + col[ni]] = fmaxf(val, 0.f);
            }
        }
    }
}

// ---------------------------------------------------------------------------
__global__ void __launch_bounds__(256)
k_pool(const float* __restrict__ H, float* __restrict__ P, int co)
{
    int o = blockIdx.x >> 5;
    int n = blockIdx.x & 31;
    const float* src = H + (size_t)o * S_TOT + (size_t)n * (T_LEN * V_NOD);
    float s = 0.f;
    for (int i = threadIdx.x; i < T_LEN * V_NOD; i += blockDim.x) s += src[i];
    __shared__ float red[256];
    red[threadIdx.x] = s;
    __syncthreads();
    for (int st = 128; st > 0; st >>= 1) {
        if (threadIdx.x < st) red[threadIdx.x] += red[threadIdx.x + st];
        __syncthreads();
    }
    if (threadIdx.x == 0)
        P[n * co + o] = red[0] * (1.f / (T_LEN * V_NOD));
}

__global__ void __launch_bounds__(256)
k_fc(const float* __restrict__ P, const float* __restrict__ W,
     const float* __restrict__ bias, float* __restrict__ O,
     int in_f, int out_f, int doRelu)
{
    int idx = blockIdx.x * blockDim.x + threadIdx.x;
    if (idx >= N_BAT * out_f) return;
    int n = idx / out_f, j = idx % out_f;
    float s = bias[j];
    for (int c = 0; c < in_f; ++c) s += P[n * in_f + c] * W[j * in_f + c];
    O[idx] = doRelu ? fmaxf(s, 0.f) : s;
}

// ---------------------------------------------------------------------------
extern "C" void kernel_launch(void* const* d_in, const int* in_sizes, int n_in,
                              void* d_out, int out_size, void* d_ws, size_t ws_size,
                              hipStream_t stream)
{
    (void)in_sizes; (void)n_in; (void)out_size; (void)ws_size;
    const float** in = (const float**)d_in;

    struct Blk {
        int ci, co, cir, hasRes, hasIdent;
        int gw, gb, b1, tw, tb, b2, rw, rb, rbn;
    };
    const Blk BLK[4] = {
        {  3,  64,   3, 1, 0,  5,  6,  7, 11, 12, 13, 17, 18, 19},
        { 64,  64,  64, 0, 1, 23, 24, 25, 29, 30, 31, -1, -1, -1},
        { 64, 128,  64, 1, 0, 35, 36, 37, 41, 42, 43, 47, 48, 49},
        {128, 256, 128, 1, 0, 53, 54, 55, 59, 60, 61, 65, 66, 67},
    };
    const int I_X = 0, I_DBN = 1;
    const int I_FC1W = 71, I_FC1B = 72, I_FC2W = 73, I_FC2B = 74, I_A = 75;

    float* wsf    = (float*)d_ws;
    float* As     = wsf;                          // 25 floats
    float* pooled = wsf + 64;                     // 32*256
    float* h1     = wsf + 64 + 8192;              // 32*128
    float* WgBuf  = wsf + 16384;                  // <= 256*128
    float* WtBuf  = wsf + 65536;                  // <= 256*2432
    float* bufA   = wsf + 1048576;                // 256 x S
    float* bufB   = bufA + (size_t)256 * S_TOT;   // 128 x S
    float* bufC   = bufB + (size_t)128 * S_TOT;   // 256 x S (+1K slack after)

    float* hinP[4]  = {bufA, bufB, bufA, bufB};
    float* tmpP[4]  = {bufB, bufA, bufB, bufA};
    float* houtP[4] = {bufB, bufA, bufB, bufA};
    float* gP = bufC;

    k_colsum<<<1, 32, 0, stream>>>(in[I_A], As);

    {
        int total = N_BAT * 3 * T_LEN * V_NOD;
        k_databn<<<(total + 255) / 256, 256, 0, stream>>>(
            in[I_X], in[I_DBN], in[I_DBN + 1], in[I_DBN + 2], in[I_DBN + 3],
            bufA);
    }

    for (int bi = 0; bi < 4; ++bi) {
        const Blk& B = BLK[bi];
        const int Kup  = (B.ci + 3) & ~3;
        const int K2p  = B.hasRes ? ((B.cir + 3) & ~3) : 0;
        const int KupT = B.co * 9 + K2p;          // co*9 already %4==0

        // node mix on input channels (commutes with the 1x1 conv)
        {
            size_t total = (size_t)B.ci * S_TOT;
            int grid = (int)((total + 255) / 256);
            k_amix<<<grid, 256, 0, stream>>>(hinP[bi], in[I_A], tmpP[bi], B.ci);
        }
        // weight prep
        k_prep_gw<<<(B.co * Kup + 255) / 256, 256, 0, stream>>>(
            in[B.gw], WgBuf, B.co, B.ci, Kup);
        {
            const float* RW  = B.hasRes ? in[B.rw]      : in[B.tw];
            const float* RG  = B.hasRes ? in[B.rbn]     : in[B.b2];
            const float* RV  = B.hasRes ? in[B.rbn + 3] : in[B.b2 + 3];
            k_prep_tw<<<(B.co * KupT + 255) / 256, 256, 0, stream>>>(
                in[B.tw], RW, in[B.b2], in[B.b2 + 3], RG, RV,
                WtBuf, B.co, B.co, B.cir, KupT, B.hasRes);
        }
        // gcn GEMM + bn1 + relu
        {
            int waves = (B.co / 64) * (S_TOT / 64);
            k_gcn<<<waves / 8, 256, 0, stream>>>(
                WgBuf, tmpP[bi], in[B.gb],
                in[B.b1], in[B.b1 + 1], in[B.b1 + 2], in[B.b1 + 3],
                As, gP, B.co, Kup);
        }
        // tcn GEMM (+ fused residual rows) + bn2 + residual + relu
        {
            int waves = (B.co / 64) * (S_TOT / 64);
            const float* RB  = B.hasRes ? in[B.rb]      : in[B.tb];
            const float* RG0 = B.hasRes ? in[B.rbn]     : in[B.b2];
            const float* RG1 = B.hasRes ? in[B.rbn + 1] : in[B.b2 + 1];
            const float* RG2 = B.hasRes ? in[B.rbn + 2] : in[B.b2 + 2];
            const float* RG3 = B.hasRes ? in[B.rbn + 3] : in[B.b2 + 3];
            k_tcn<<<waves / 8, 256, 0, stream>>>(
                WtBuf, gP, hinP[bi], in[B.tb],
                in[B.b2], in[B.b2 + 1], in[B.b2 + 2], in[B.b2 + 3],
                RB, RG0, RG1, RG2, RG3,
                houtP[bi], B.co, B.co, KupT, K2p, B.hasRes, B.hasIdent);
        }
    }

    k_pool<<<256 * N_BAT, 256, 0, stream>>>(houtP[3], pooled, 256);
    k_fc<<<(N_BAT * 128 + 255) / 256, 256, 0, stream>>>(
        pooled, in[I_FC1W], in[I_FC1B], h1, 256, 128, 1);
    k_fc<<<(N_BAT * 128 + 255) / 256, 256, 0, stream>>>(
        h1, in[I_FC2W], in[I_FC2B], (float*)d_out, 128, 128, 0);
}